// PillarEncoder_60009283060123
// MI455X (gfx1250) — compile-verified
//
#include <hip/hip_runtime.h>

typedef float v2f __attribute__((ext_vector_type(2)));
typedef float v8f __attribute__((ext_vector_type(8)));

#define P_TOTAL 80000
#define NPTS 32
#define OUT_C 64
#define IN_C 9
#define X_L 432
#define Y_L 496
#define VOX 0.16f
#define XOFF 0.08f
#define YOFF (-39.60f)
#define BN_EPS 1e-3f

__global__ __launch_bounds__(256) void zero_f4(float* __restrict__ out, int n) {
    int i = blockIdx.x * blockDim.x + threadIdx.x;
    int stride = gridDim.x * blockDim.x;
    int n4 = n >> 2;
    float4 z = make_float4(0.f, 0.f, 0.f, 0.f);
    for (int j = i; j < n4; j += stride)
        ((float4*)out)[j] = z;
    for (int j = (n4 << 2) + i; j < n; j += stride)
        out[j] = 0.f;
}

__global__ __launch_bounds__(256) void pillar_wmma_kernel(
    const float* __restrict__ pillars,     // (P, 32, 4)
    const int*   __restrict__ npoints,     // (P,)
    const int*   __restrict__ coors,       // (P, 4) [b, x, y, 0]
    const float* __restrict__ conv_w,      // (64, 9)
    const float* __restrict__ bn_gamma,
    const float* __restrict__ bn_beta,
    const float* __restrict__ bn_mean,
    const float* __restrict__ bn_var,
    float*       __restrict__ out)         // (4, 64, 496, 432)
{
    const int lane = threadIdx.x & 31;
    const int wave = threadIdx.x >> 5;
    const int wave_global = blockIdx.x * (blockDim.x >> 5) + wave;
    const int wave_stride = gridDim.x * (blockDim.x >> 5);
    const bool lo = (lane < 16);
    const int  l15 = lane & 15;

    // ---- BN fold: scale goes into the weights, shift applied after max ----
    // y = relu((Wf)*a + b); max_i relu(a*x_i + b) = relu(max_i(a*x_i) + b)
    float bnb[4];
    float bna[4];
#pragma unroll
    for (int n = 0; n < 4; ++n) {
        int ch = n * 16 + l15;
        float s = bn_gamma[ch] / sqrtf(bn_var[ch] + BN_EPS);
        bna[n] = s;
        bnb[n] = bn_beta[ch] - bn_mean[ch] * s;
    }

    // ---- B fragments (weights^T * bn_scale, padded K 9->12), loop-invariant ----
    // B layout (4x16 f32): v0: lanes0-15 = row 4k+0, lanes16-31 = row 4k+2
    //                      v1: lanes0-15 = row 4k+1, lanes16-31 = row 4k+3
    // Each lane's B column == one output channel, so scaling the fragment by
    // bna[n] scales exactly that channel's matmul output.
    v2f wf[4][3];
#pragma unroll
    for (int n = 0; n < 4; ++n) {
        int col = n * 16 + l15;  // output channel
#pragma unroll
        for (int k = 0; k < 3; ++k) {
            int r0 = 4 * k + (lo ? 0 : 2);
            int r1 = r0 + 1;
            v2f b;
            b.x = (r0 < IN_C) ? conv_w[col * IN_C + r0] * bna[n] : 0.f;
            b.y = (r1 < IN_C) ? conv_w[col * IN_C + r1] * bna[n] : 0.f;
            wf[n][k] = b;
        }
    }

    for (int p = wave_global; p < P_TOTAL; p += wave_stride) {
        float4 pt = ((const float4*)pillars)[p * NPTS + lane];  // lane = point idx
        int npts = npoints[p];
        int cb = coors[p * 4 + 0];
        int cx = coors[p * 4 + 1];
        int cy = coors[p * 4 + 2];

        // full 32-point sums (reference sums ALL points, divides by npoints)
        float sx = pt.x, sy = pt.y, sz = pt.z;
#pragma unroll
        for (int off = 16; off >= 1; off >>= 1) {
            sx += __shfl_xor(sx, off, 32);
            sy += __shfl_xor(sy, off, 32);
            sz += __shfl_xor(sz, off, 32);
        }
        float inv_n = 1.0f / (float)npts;
        float mx = sx * inv_n, my = sy * inv_n, mz = sz * inv_n;

        float f[IN_C];
        f[0] = pt.x; f[1] = pt.y; f[2] = pt.z; f[3] = pt.w;
        f[4] = pt.x - mx; f[5] = pt.y - my; f[6] = pt.z - mz;
        f[7] = pt.x - ((float)cx * VOX + XOFF);
        f[8] = pt.y - ((float)cy * VOX + YOFF);
        float msk = (lane < npts) ? 1.f : 0.f;
#pragma unroll
        for (int c = 0; c < IN_C; ++c) f[c] *= msk;

        // ---- A fragments via cross-lane gather (16x4 f32 A layout) ----
        // lane L<16 needs cols 4k,4k+1 of point m*16+L; lane L>=16 cols 4k+2,4k+3
        v2f af[2][3];
#pragma unroll
        for (int m = 0; m < 2; ++m) {
            int s = m * 16 + l15;  // source lane == point index
            {   // k=0: cols 0..3
                float g0 = __shfl(f[0], s, 32), g1 = __shfl(f[1], s, 32);
                float g2 = __shfl(f[2], s, 32), g3 = __shfl(f[3], s, 32);
                v2f a; a.x = lo ? g0 : g2; a.y = lo ? g1 : g3; af[m][0] = a;
            }
            {   // k=1: cols 4..7
                float g0 = __shfl(f[4], s, 32), g1 = __shfl(f[5], s, 32);
                float g2 = __shfl(f[6], s, 32), g3 = __shfl(f[7], s, 32);
                v2f a; a.x = lo ? g0 : g2; a.y = lo ? g1 : g3; af[m][1] = a;
            }
            {   // k=2: cols 8..11 (9..11 zero pad)
                float g0 = __shfl(f[8], s, 32);
                v2f a; a.x = lo ? g0 : 0.f; a.y = 0.f; af[m][2] = a;
            }
        }

        // ---- 24x V_WMMA_F32_16X16X4_F32: acc[m][n] = sum_k A[m][k] B[k][n] ----
        v8f acc[2][4];
#pragma unroll
        for (int m = 0; m < 2; ++m)
#pragma unroll
            for (int n = 0; n < 4; ++n) {
                v8f c = {};
#pragma unroll
                for (int k = 0; k < 3; ++k)
                    c = __builtin_amdgcn_wmma_f32_16x16x4_f32(
                        false, af[m][k], false, wf[n][k], (short)0, c, false, false);
                acc[m][n] = c;
            }

        // ---- max over 32 points (values already BN-scaled), shift+ReLU, scatter ----
        size_t plane = (size_t)Y_L * X_L;
        size_t base = (size_t)cb * OUT_C * plane + (size_t)cy * X_L + (size_t)cx;
#pragma unroll
        for (int n = 0; n < 4; ++n) {
            float vmax = fmaxf(acc[0][n][0], acc[1][n][0]);
#pragma unroll
            for (int g = 1; g < 8; ++g)
                vmax = fmaxf(vmax, fmaxf(acc[0][n][g], acc[1][n][g]));
            vmax = fmaxf(vmax, __shfl_xor(vmax, 16, 32));  // merge M halves
            float val = fmaxf(vmax + bnb[n], 0.f);          // BN shift + ReLU
            // lanes 0-15 store n=0,1; lanes 16-31 store n=2,3 (2 stores/lane)
            if (lo ? (n < 2) : (n >= 2)) {
                int ch = n * 16 + l15;
                out[base + (size_t)ch * plane] = val;
            }
        }
    }
}

extern "C" void kernel_launch(void* const* d_in, const int* in_sizes, int n_in,
                              void* d_out, int out_size, void* d_ws, size_t ws_size,
                              hipStream_t stream) {
    const float* pillars  = (const float*)d_in[0];
    const int*   npoints  = (const int*)d_in[1];
    const int*   coors    = (const int*)d_in[2];
    const float* conv_w   = (const float*)d_in[3];
    const float* bn_gamma = (const float*)d_in[4];
    const float* bn_beta  = (const float*)d_in[5];
    const float* bn_mean  = (const float*)d_in[6];
    const float* bn_var   = (const float*)d_in[7];
    float* out = (float*)d_out;

    // canvas is mostly zeros; must be cleared deterministically every call
    zero_f4<<<2048, 256, 0, stream>>>(out, out_size);
    pillar_wmma_kernel<<<1024, 256, 0, stream>>>(
        pillars, npoints, coors, conv_w, bn_gamma, bn_beta, bn_mean, bn_var, out);
}